// VF_27195732918308
// MI455X (gfx1250) — compile-verified
//
#include <hip/hip_runtime.h>

typedef __attribute__((ext_vector_type(2))) float v2f;
typedef __attribute__((ext_vector_type(8))) float v8f;

#define IN_CH  4
#define HIDDEN 32
#define NNODES 4

static __device__ __forceinline__ v8f wmma_f32_4(v2f a, v2f b, v8f c) {
    // D = A(16x4) * B(4x16) + C(16x16), fp32
    return __builtin_amdgcn_wmma_f32_16x16x4_f32(false, a, false, b, (short)0, c, false, false);
}

// ---------------- K1: init degree to 1 (self loop) ----------------
__global__ void k_init_deg(unsigned* __restrict__ deg, int n) {
    int i = blockIdx.x * blockDim.x + threadIdx.x;
    if (i < n) deg[i] = 1u;
}

// ---------------- K2: degree count over edge destinations ----------------
__global__ void k_degree(const int* __restrict__ dst, unsigned* __restrict__ deg, int nE) {
    int e = blockIdx.x * blockDim.x + threadIdx.x;
    if (e < nE) atomicAdd(&deg[dst[e]], 1u);
}

// ---------------- K3: dinv = rsqrt(deg); pooled = sum4(xw/deg) (self loops) ----------------
__global__ void k_self(const float* __restrict__ state,
                       const float* __restrict__ Wg,     // [4,32] row-major
                       const unsigned* __restrict__ deg,
                       float* __restrict__ dinv,
                       float* __restrict__ pooled,       // [N/4, 32]
                       int nNodes) {
    const int lane = threadIdx.x & 31;
    const int wid  = (blockIdx.x * blockDim.x + threadIdx.x) >> 5;
    const int nbase = wid * 16;
    if (nbase + 16 > nNodes) return;                 // wave-uniform

    const int m  = lane & 15;                        // row (A) / col (B,C)
    const int kh = lane >> 4;                        // K-half select
    const int k0 = kh * 2;

    // A: 16 node rows of state (16x4)
    v2f a;
    { const float* sp = state + (size_t)(nbase + m) * IN_CH + k0;
      a.x = sp[0]; a.y = sp[1]; }
    // B tiles: W_gcn columns [0..15] and [16..31]
    v2f b0, b1;
    b0.x = Wg[(k0 + 0) * HIDDEN + m];
    b0.y = Wg[(k0 + 1) * HIDDEN + m];
    b1.x = Wg[(k0 + 0) * HIDDEN + 16 + m];
    b1.y = Wg[(k0 + 1) * HIDDEN + 16 + m];

    v8f c0 = {}, c1 = {};
    c0 = wmma_f32_4(a, b0, c0);
    c1 = wmma_f32_4(a, b1, c1);

    // Scale row r by 1/deg (self-loop norm = dinv^2), sum groups of 4 rows -> pooled rows.
    // deg >= 1 always (self loops), so the fast hardware reciprocal is safe.
    float q0 = 0.f, q1 = 0.f, q2 = 0.f, q3 = 0.f;
#pragma unroll
    for (int r = 0; r < 8; ++r) {
        int row = kh * 8 + r;
        float inv = __builtin_amdgcn_rcpf((float)deg[nbase + row]);   // v_rcp_f32
        float e0 = c0[r] * inv;
        float e1 = c1[r] * inv;
        if (r < 4) { q0 += e0; q2 += e1; } else { q1 += e0; q3 += e1; }
    }

    if (lane < 16)
        dinv[nbase + lane] = __builtin_amdgcn_rsqf((float)deg[nbase + lane]); // v_rsq_f32

    const int bb = (nbase >> 2) + kh * 2;            // pooled group index
    pooled[(size_t)bb * HIDDEN + m]            = q0;
    pooled[(size_t)bb * HIDDEN + 16 + m]       = q2;
    pooled[(size_t)(bb + 1) * HIDDEN + m]      = q1;
    pooled[(size_t)(bb + 1) * HIDDEN + 16 + m] = q3;
}

// ---------------- K4: edge scatter (wave = 16 edges, 2 WMMAs, f32 atomics) ----------------
__global__ void k_edges(const int* __restrict__ src, const int* __restrict__ dst,
                        const float* __restrict__ state,
                        const float* __restrict__ Wg,
                        const float* __restrict__ dinv,
                        float* __restrict__ pooled, int nE) {
    const int lane = threadIdx.x & 31;
    const int wid  = (blockIdx.x * blockDim.x + threadIdx.x) >> 5;
    const int ebase = wid * 16;
    if (ebase + 16 > nE) return;                     // wave-uniform

    const int m  = lane & 15;
    const int kh = lane >> 4;
    const int k0 = kh * 2;

    const int e = ebase + m;
    const int s = src[e];
    const int d = dst[e];
    const float nrm = dinv[s] * dinv[d];

    // A: norm-scaled source states (16x4); state (64 MB) stays L2-resident
    v2f a;
    { const float* sp = state + (size_t)s * IN_CH + k0;
      a.x = sp[0] * nrm; a.y = sp[1] * nrm; }
    v2f b0, b1;
    b0.x = Wg[(k0 + 0) * HIDDEN + m];
    b0.y = Wg[(k0 + 1) * HIDDEN + m];
    b1.x = Wg[(k0 + 0) * HIDDEN + 16 + m];
    b1.y = Wg[(k0 + 1) * HIDDEN + 16 + m];

    v8f c0 = {}, c1 = {};
    c0 = wmma_f32_4(a, b0, c0);
    c1 = wmma_f32_4(a, b1, c1);

    const int bidx = d >> 2;                         // scatter straight into pooled group
#pragma unroll
    for (int r = 0; r < 8; ++r) {
        int row = kh * 8 + r;                        // edge whose message row this lane holds
        int bd  = __shfl(bidx, row, 32);
        float* p = pooled + (size_t)bd * HIDDEN + m;
        atomicAdd(p,      c0[r]);
        atomicAdd(p + 16, c1[r]);
    }
}

// ---------------- K5: build v=[sum4(state), pooled(+4*b_gcn)], MLP via chained WMMA ----------------
__global__ void k_mlp(const float* __restrict__ state,
                      const float* __restrict__ pooled,
                      const float* __restrict__ bg,    // b_gcn [32]
                      const float* __restrict__ W1,    // [36,32]
                      const float* __restrict__ b1,    // [32]
                      const float* __restrict__ W2,    // [32]
                      const float* __restrict__ b2v,   // [1]
                      float* __restrict__ out, int nB) {
    __shared__ float lds[8 * 16 * 36];               // 8 waves * (16x36) v-tile
    const int lane   = threadIdx.x & 31;
    const int wlocal = threadIdx.x >> 5;
    const int wid    = (blockIdx.x * blockDim.x + threadIdx.x) >> 5;
    const int bbase  = wid * 16;
    if (bbase + 16 > nB) return;                     // wave-uniform

    float* vt = &lds[wlocal * 16 * 36];

    // Build 16x36 v-tile: chunk 0 = sum of 4 node states, chunks 1..8 = pooled (+ NNODES*b_gcn)
    for (int idx = lane; idx < 16 * 9; idx += 32) {
        int row = idx / 9, ch = idx % 9;
        float4 val;
        if (ch == 0) {
            const float4* sp = (const float4*)(state + (size_t)(bbase + row) * NNODES * IN_CH);
            float4 s0 = sp[0], s1 = sp[1], s2 = sp[2], s3 = sp[3];
            val.x = s0.x + s1.x + s2.x + s3.x;
            val.y = s0.y + s1.y + s2.y + s3.y;
            val.z = s0.z + s1.z + s2.z + s3.z;
            val.w = s0.w + s1.w + s2.w + s3.w;
        } else {
            val = ((const float4*)(pooled + (size_t)(bbase + row) * HIDDEN))[ch - 1];
            float4 g = ((const float4*)bg)[ch - 1];
            val.x += (float)NNODES * g.x;
            val.y += (float)NNODES * g.y;
            val.z += (float)NNODES * g.z;
            val.w += (float)NNODES * g.w;
        }
        ((float4*)vt)[row * 9 + ch] = val;
    }
    __syncthreads();

    const int m  = lane & 15;
    const int kh = lane >> 4;
    v8f c0 = {}, c1 = {};
#pragma unroll
    for (int t = 0; t < 9; ++t) {
        int k0 = t * 4 + kh * 2;
        v2f a;  a.x = vt[m * 36 + k0]; a.y = vt[m * 36 + k0 + 1];
        v2f w0, w1;
        w0.x = W1[(k0 + 0) * HIDDEN + m];
        w0.y = W1[(k0 + 1) * HIDDEN + m];
        w1.x = W1[(k0 + 0) * HIDDEN + 16 + m];
        w1.y = W1[(k0 + 1) * HIDDEN + 16 + m];
        c0 = wmma_f32_4(a, w0, c0);
        c1 = wmma_f32_4(a, w1, c1);
    }

    // bias + ReLU + layer-2 partial dot per lane
    const float b1a = b1[m],      b1b = b1[16 + m];
    const float w2a = W2[m],      w2b = W2[16 + m];
    float p[8];
#pragma unroll
    for (int r = 0; r < 8; ++r) {
        float h0 = fmaxf(c0[r] + b1a, 0.0f);
        float h1 = fmaxf(c1[r] + b1b, 0.0f);
        p[r] = h0 * w2a + h1 * w2b;
    }
    // reduce across the 16 lanes of each half (xor stays within the half)
#pragma unroll
    for (int off = 1; off < 16; off <<= 1) {
#pragma unroll
        for (int r = 0; r < 8; ++r) p[r] += __shfl_xor(p[r], off, 32);
    }
    if (m == 0) {
        float bias = b2v[0];
        int ob = bbase + kh * 8;
#pragma unroll
        for (int r = 0; r < 8; ++r) out[ob + r] = p[r] + bias;
    }
}

extern "C" void kernel_launch(void* const* d_in, const int* in_sizes, int n_in,
                              void* d_out, int out_size, void* d_ws, size_t ws_size,
                              hipStream_t stream) {
    const float* state = (const float*)d_in[0];
    const int*   eidx  = (const int*)d_in[1];        // [2, E] flat
    const float* Wg    = (const float*)d_in[2];
    const float* bg    = (const float*)d_in[3];
    const float* W1    = (const float*)d_in[4];
    const float* b1    = (const float*)d_in[5];
    const float* W2    = (const float*)d_in[6];
    const float* b2    = (const float*)d_in[7];
    float* out = (float*)d_out;

    const int N = in_sizes[0] / IN_CH;               // 4,000,000
    const int E = in_sizes[1] / 2;                   // 12,000,000
    const int nB = N / NNODES;                       // 1,000,000
    const int* src = eidx;
    const int* dst = eidx + E;

    // Workspace layout: deg u32[N] | dinv f32[N] | pooled f32[nB*32]  (~160 MB)
    unsigned* deg   = (unsigned*)d_ws;
    float*    dinv  = (float*)((char*)d_ws + (size_t)N * 4);
    float*    pooled= (float*)((char*)d_ws + (size_t)N * 8);

    const int BLK = 256;                             // 8 waves/block (wave32)

    // K1: deg = 1
    k_init_deg<<<(N + BLK - 1) / BLK, BLK, 0, stream>>>(deg, N);
    // K2: deg += indegree
    k_degree<<<(E + BLK - 1) / BLK, BLK, 0, stream>>>(dst, deg, E);
    // K3: dinv + self-loop contribution into pooled (16 nodes / wave)
    {
        int waves = N / 16;
        k_self<<<waves / 8, BLK, 0, stream>>>(state, Wg, deg, dinv, pooled, N);
    }
    // K4: edge scatter (16 edges / wave)
    {
        int waves = E / 16;
        k_edges<<<waves / 8, BLK, 0, stream>>>(src, dst, state, Wg, dinv, pooled, E);
    }
    // K5: pooled MLP (16 outputs / wave)
    {
        int waves = nB / 16;
        k_mlp<<<waves / 8, BLK, 0, stream>>>(state, pooled, bg, W1, b1, W2, b2, out, nB);
    }
}